// Health_State_Analysis_60224031425109
// MI455X (gfx1250) — compile-verified
//
#include <hip/hip_runtime.h>
#include <stdint.h>

#ifndef __has_builtin
#define __has_builtin(x) 0
#endif

#define S_LEN   2560
#define THREADS 320          // 10 waves of 32 (wave32 on gfx1250)
#define NWAVES  (THREADS / 32)
#define PT      8            // elements per thread: 320*8 = 2560
#define CPOL_NT 1            // gfx12 CPol TH_NT: streamed once, don't retain in cache

// ---- wave32 butterfly reductions ----
__device__ __forceinline__ float wsum(float v) {
#pragma unroll
  for (int o = 16; o > 0; o >>= 1) v += __shfl_xor(v, o, 32);
  return v;
}
__device__ __forceinline__ float wrmax(float v) {
#pragma unroll
  for (int o = 16; o > 0; o >>= 1) v = fmaxf(v, __shfl_xor(v, o, 32));
  return v;
}
__device__ __forceinline__ float wrmin(float v) {
#pragma unroll
  for (int o = 16; o > 0; o >>= 1) v = fminf(v, __shfl_xor(v, o, 32));
  return v;
}

__device__ __forceinline__ int sgnf(float v) { return (v > 0.f) - (v < 0.f); }

typedef int v4i __attribute__((ext_vector_type(4)));
typedef __attribute__((address_space(1))) v4i gv4i;   // global int4
typedef __attribute__((address_space(3))) v4i lv4i;   // LDS int4

__global__ __launch_bounds__(THREADS) void Health_State_Analysis_kernel(
    const float* __restrict__ in, float* __restrict__ out, int nrows) {
  const int row  = blockIdx.x;
  if (row >= nrows) return;
  const int tid  = threadIdx.x;
  const int lane = tid & 31;
  const int wv   = tid >> 5;

  __shared__ float s_x[S_LEN];              // staged row (10 KB of 320 KB/WGP)
  __shared__ float s_part[NWAVES][16];      // per-wave partials
  __shared__ float s_ms[2];                 // mean, std broadcast

  const float* rp  = in + (size_t)row * S_LEN;
  float*       rpm = const_cast<float*>(rp);

  // ---- stage row into LDS via gfx1250 async global->LDS (ASYNCcnt path) ----
  {
    const int j0 = tid * 4;                 // first float4 of this lane
    const int j1 = (tid + THREADS) * 4;     // second float4 of this lane
#if __has_builtin(__builtin_amdgcn_global_load_async_to_lds_b128)
    __builtin_amdgcn_global_load_async_to_lds_b128((gv4i*)(rpm + j0),
                                                   (lv4i*)&s_x[j0], 0, CPOL_NT);
    __builtin_amdgcn_global_load_async_to_lds_b128((gv4i*)(rpm + j1),
                                                   (lv4i*)&s_x[j1], 0, CPOL_NT);
#else
    {
      uint32_t l0 = (uint32_t)(uintptr_t)&s_x[j0];   // raw LDS byte offset
      uint32_t l1 = (uint32_t)(uintptr_t)&s_x[j1];
      const float* g0 = rp + j0;
      const float* g1 = rp + j1;
      asm volatile("global_load_async_to_lds_b128 %0, %1, off"
                   :: "v"(l0), "v"(g0) : "memory");
      asm volatile("global_load_async_to_lds_b128 %0, %1, off"
                   :: "v"(l1), "v"(g1) : "memory");
    }
#endif
#if __has_builtin(__builtin_amdgcn_s_wait_asynccnt)
    __builtin_amdgcn_s_wait_asynccnt(0);
#else
    asm volatile("s_wait_asynccnt 0" ::: "memory");
#endif
  }
  __syncthreads();

  // ---- pass 1: per-thread contiguous chunk [8t, 8t+8) + 2 neighbors ----
  const int base = tid * PT;
  float x[PT + 2];
  {
    float4 a = *(const float4*)&s_x[base];
    float4 b = *(const float4*)&s_x[base + 4];
    x[0] = a.x; x[1] = a.y; x[2] = a.z; x[3] = a.w;
    x[4] = b.x; x[5] = b.y; x[6] = b.z; x[7] = b.w;
    // branch-free clamped neighbor loads; clamped values are never consumed
    // (the gi < S_LEN-1 / S_LEN-2 guards exclude them from every accumulation)
    const int c8 = (base + 8 < S_LEN) ? base + 8 : S_LEN - 1;
    const int c9 = (base + 9 < S_LEN) ? base + 9 : S_LEN - 1;
    x[8] = s_x[c8];
    x[9] = s_x[c9];
  }

  float s1 = 0.f, s2 = 0.f, s3 = 0.f, s4 = 0.f;
  float mx = -INFINITY, mn = INFINITY, asum = 0.f, amax = 0.f;
  float d1s = 0.f, d1q = 0.f, d2s = 0.f, d2q = 0.f;
  int zc = 0;
#pragma unroll
  for (int i = 0; i < PT; ++i) {
    const float v  = x[i];
    const float v2 = v * v;
    s1 += v;
    s2 = fmaf(v, v, s2);
    s3 = fmaf(v2, v, s3);
    s4 = fmaf(v2, v2, s4);
    mx = fmaxf(mx, v);
    mn = fminf(mn, v);
    const float av = fabsf(v);
    asum += av;
    amax = fmaxf(amax, av);
    const int gi = base + i;
    if (gi < S_LEN - 1) {
      const float d = x[i + 1] - v;
      d1s += d;
      d1q = fmaf(d, d, d1q);
      zc += (sgnf(x[i + 1]) != sgnf(v));
    }
    if (gi < S_LEN - 2) {
      const float e = x[i + 2] - 2.f * x[i + 1] + v;
      d2s += e;
      d2q = fmaf(e, e, d2q);
    }
  }

  // wave-level reduce, per-wave partials to LDS
  s1 = wsum(s1);  s2 = wsum(s2);  s3 = wsum(s3);  s4 = wsum(s4);
  asum = wsum(asum);
  float zcf = wsum((float)zc);
  d1s = wsum(d1s); d1q = wsum(d1q); d2s = wsum(d2s); d2q = wsum(d2q);
  mx = wrmax(mx);  mn = wrmin(mn);  amax = wrmax(amax);
  if (lane == 0) {
    float* p = s_part[wv];
    p[0] = s1;  p[1] = s2;  p[2] = s3;  p[3] = s4;  p[4] = asum;
    p[5] = zcf; p[6] = d1s; p[7] = d1q; p[8] = d2s; p[9] = d2q;
    p[10] = mx; p[11] = mn; p[12] = amax;
  }
  __syncthreads();

  // thread 0: fold waves, derive mean/std, broadcast
  float mean = 0.f, varu = 0.f, stdv = 0.f, rms = 0.f, skew = 0.f, kurt = 0.f;
  float shape = 0.f, impulse = 0.f, zcr = 0.f, mob = 0.f, comp = 0.f;
  float MX = 0.f, MN = 0.f;
  if (tid == 0) {
    float S1 = 0.f, S2 = 0.f, S3 = 0.f, S4 = 0.f, AS = 0.f, ZC = 0.f;
    float D1 = 0.f, D1Q = 0.f, D2 = 0.f, D2Q = 0.f;
    MX = -INFINITY; MN = INFINITY;
    float AM = 0.f;
#pragma unroll
    for (int w = 0; w < NWAVES; ++w) {
      const float* p = s_part[w];
      S1 += p[0]; S2 += p[1]; S3 += p[2]; S4 += p[3]; AS += p[4];
      ZC += p[5]; D1 += p[6]; D1Q += p[7]; D2 += p[8]; D2Q += p[9];
      MX = fmaxf(MX, p[10]); MN = fminf(MN, p[11]); AM = fmaxf(AM, p[12]);
    }
    const float n = (float)S_LEN;
    mean = S1 / n;
    varu = (S2 - S1 * S1 / n) / (n - 1.f);       // ddof=1
    stdv = sqrtf(varu);
    rms  = sqrtf(S2 / n);
    const float m2r = S2 / n, m3r = S3 / n, m4r = S4 / n;
    const float m3 = m3r - 3.f * mean * m2r + 2.f * mean * mean * mean;
    const float m4 = m4r - 4.f * mean * m3r + 6.f * mean * mean * m2r
                     - 3.f * mean * mean * mean * mean;
    skew = m3 / (stdv * stdv * stdv);
    kurt = m4 / (varu * varu);                    // std^4
    shape   = rms * n / AS;
    impulse = AM * n / AS;
    zcr = ZC / (2.f * n);
    const float n1 = (float)(S_LEN - 1), n2 = (float)(S_LEN - 2);
    const float vd1 = (D1Q - D1 * D1 / n1) / (n1 - 1.f);
    const float vd2 = (D2Q - D2 * D2 / n2) / (n2 - 1.f);
    mob  = sqrtf(vd1 / varu);
    comp = sqrtf(vd2 / vd1);
    s_ms[0] = mean;
    s_ms[1] = stdv;
  }
  __syncthreads();

  // ---- pass 2: outliers |x - mean| > 3*std (row data still in VGPRs) ----
  {
    const float mu = s_ms[0];
    const float thr = 3.0f * s_ms[1];
    int c = 0;
#pragma unroll
    for (int i = 0; i < PT; ++i) c += (fabsf(x[i] - mu) > thr) ? 1 : 0;
    const float cf = wsum((float)c);
    if (lane == 0) s_part[wv][0] = cf;
  }
  __syncthreads();

  if (tid == 0) {
    float OUTL = 0.f;
#pragma unroll
    for (int w = 0; w < NWAVES; ++w) OUTL += s_part[w][0];
    float* o = out + (size_t)row * 15;
    o[0]  = mean;
    o[1]  = MX;
    o[2]  = MN;
    o[3]  = MX - MN;
    o[4]  = varu;        // var
    o[5]  = rms;
    o[6]  = skew;
    o[7]  = kurt;
    o[8]  = shape;
    o[9]  = impulse;
    o[10] = OUTL;
    o[11] = zcr;
    o[12] = varu;        // activity
    o[13] = mob;
    o[14] = comp;
  }
}

extern "C" void kernel_launch(void* const* d_in, const int* in_sizes, int n_in,
                              void* d_out, int out_size, void* d_ws, size_t ws_size,
                              hipStream_t stream) {
  (void)n_in; (void)out_size; (void)d_ws; (void)ws_size;
  const float* in = (const float*)d_in[0];
  float* out = (float*)d_out;
  const int nrows = in_sizes[0] / S_LEN;   // 16384 rows of 2560 floats
  Health_State_Analysis_kernel<<<dim3(nrows), dim3(THREADS), 0, stream>>>(in, out, nrows);
}